// Yang19_CTRNNModel_34952443855108
// MI455X (gfx1250) — compile-verified
//
#include <hip/hip_runtime.h>

// Problem dims (compile-time constants from the reference)
#define SEQ      1024
#define BATCH    128
#define IN_DIM   128
#define HID      512
#define OUT_DIM  64
#define MTOT     (SEQ * BATCH)          // 131072 flattened (s,b) rows
#define ALPHA_F  0.2f
#define SIGMA_F  0.031622776601683794f  // sqrt(2*0.2)*0.05

#define NBLK_SCAN 8                      // persistent blocks for the scan

typedef __attribute__((ext_vector_type(2))) float v2f;
typedef __attribute__((ext_vector_type(8))) float v8f;
typedef __attribute__((ext_vector_type(2))) int   v2i;

// Pointer types matching the async-to-LDS builtin's parameters:
// arg0: global (AS1) pointer to 64-bit payload, arg1: LDS (AS3) pointer.
typedef __attribute__((address_space(1))) v2i* gbl_b64_ptr;
typedef __attribute__((address_space(3))) v2i* lds_b64_ptr;

// Probe for gfx1250 async global->LDS copy builtins (ASYNCcnt path).
#if defined(__has_builtin)
#  if __has_builtin(__builtin_amdgcn_global_load_async_to_lds_b64) && \
      __has_builtin(__builtin_amdgcn_s_wait_asynccnt)
#    define USE_ASYNC_LDS 1
#  endif
#endif
#ifndef USE_ASYNC_LDS
#  define USE_ASYNC_LDS 0
#endif

// ---------------------------------------------------------------------------
// V_WMMA_F32_16X16X4_F32 wrapper.  D = A(16x4) * B(4x16) + C(16x16), fp32.
// ---------------------------------------------------------------------------
__device__ __forceinline__ v8f wmma4(v2f a, v2f b, v8f c) {
    return __builtin_amdgcn_wmma_f32_16x16x4_f32(
        false, a, false, b, (short)0, c, false, false);
}

// A-fragment loader from a row-major M x K matrix in global memory.
// lane holds row (rc0 + lane&15), K = kk + 2*(lane>>4) .. +1  -> one b64 load.
__device__ __forceinline__ v2f load_frag(const float* __restrict__ p,
                                         int rc0, int ld, int kk, int lane) {
    const float* q = p + (size_t)(rc0 + (lane & 15)) * ld
                       + kk + ((lane >> 4) << 1);
    v2f r;
    r.x = q[0];
    r.y = q[1];
    return r;
}

// ---------------------------------------------------------------------------
// Weight staging: pre-swizzle W (row-major N x K) into LDS as ready-to-issue
// B fragments.  Fragment (cg, kki) = 32 lanes x 2 floats contiguous (256 B):
//   lds[((cg*nk4 + kki)*32 + lane)*2 + j] = W[(col0+cg*16+(lane&15))*ldw
//                                             + kki*4 + 2*(lane>>4) + j]
// Reading a fragment is then a single conflict-free ds_load_b64 per lane.
// Fill uses the gfx1250 async global->LDS path when available (each lane
// supplies both addresses, so the swizzle is preserved).
// ---------------------------------------------------------------------------
__device__ __forceinline__ void fill_wfrags(float* lds, const float* __restrict__ W,
                                            int col0, int ldw, int ncg, int nk4,
                                            int tid, int nthreads) {
    const int total = ncg * nk4 * 32;     // fragment-lane slots
    for (int slot = tid; slot < total; slot += nthreads) {
        const int l   = slot & 31;
        const int kki = (slot >> 5) % nk4;
        const int cg  = (slot >> 5) / nk4;
        const int n = col0 + cg * 16 + (l & 15);
        const int k = kki * 4 + ((l >> 4) << 1);
        const float* src = W + (size_t)n * ldw + k;
        float* dst = lds + (size_t)slot * 2;
#if USE_ASYNC_LDS
        __builtin_amdgcn_global_load_async_to_lds_b64(
            (gbl_b64_ptr)src, (lds_b64_ptr)dst, 0, 0);
#else
        dst[0] = src[0];
        dst[1] = src[1];
#endif
    }
#if USE_ASYNC_LDS
    __builtin_amdgcn_s_wait_asynccnt(0);
#endif
    __syncthreads();
}

__device__ __forceinline__ v2f lds_frag(const float* lds, int cg, int nk4,
                                        int kki, int lane) {
    const float* q = lds + ((size_t)(cg * nk4 + kki) * 32 + lane) * 2;
    v2f r;
    r.x = q[0];
    r.y = q[1];
    return r;
}

// ---------------------------------------------------------------------------
// Kernel 1: proj_plus[s,b,h] = x[s,b,:]·W_in[h,:] + b_in[h] + b_h[h]
//                              + SIGMA*noise[s,b,h]    (into rnn region)
// Block tile 128 x 64, 8 waves; wave w = row group w, 4 col subtiles.
// W_in slice staged in 32 KB LDS.  Grid: (1024, 8).
// ---------------------------------------------------------------------------
__global__ __launch_bounds__(256)
void inproj_kernel(const float* __restrict__ x, const float* __restrict__ noise,
                   const float* __restrict__ Win, const float* __restrict__ bin,
                   const float* __restrict__ bh, float* __restrict__ proj) {
    extern __shared__ float smem[];
    const int lane = threadIdx.x & 31;
    const int wave = threadIdx.x >> 5;                 // 0..7
    const int rowbase = blockIdx.x * 128 + wave * 16;  // flattened (s,b) row
    const int colblk  = blockIdx.y * 64;
    const int NK4 = IN_DIM / 4;                        // 32

    fill_wfrags(smem, Win, colblk, IN_DIM, 4, NK4, threadIdx.x, 256);

    v8f acc[4];
    float bias[4];
#pragma unroll
    for (int cg = 0; cg < 4; ++cg) {
        const int n = colblk + cg * 16 + (lane & 15);
        bias[cg] = bin[n] + bh[n];
#pragma unroll
        for (int v = 0; v < 8; ++v) acc[cg][v] = 0.0f;
    }

    for (int kki = 0; kki < NK4; kki += 4) {           // chunk of 4 fragments
        v2f a[4], b[4][4];
#pragma unroll
        for (int j = 0; j < 4; ++j) {
            a[j] = load_frag(x, rowbase, IN_DIM, (kki + j) * 4, lane);
#pragma unroll
            for (int cg = 0; cg < 4; ++cg)
                b[j][cg] = lds_frag(smem, cg, NK4, kki + j, lane);
        }
#pragma unroll
        for (int j = 0; j < 4; ++j)
#pragma unroll
            for (int cg = 0; cg < 4; ++cg)
                acc[cg] = wmma4(a[j], b[j][cg], acc[cg]);
    }

#pragma unroll
    for (int cg = 0; cg < 4; ++cg) {
        const int n = colblk + cg * 16 + (lane & 15);
#pragma unroll
        for (int v = 0; v < 8; ++v) {
            const int m = rowbase + ((lane >> 4) << 3) + v;
            const size_t idx = (size_t)m * HID + n;
            proj[idx] = acc[cg][v] + bias[cg] + SIGMA_F * noise[idx];
        }
    }
}

// ---------------------------------------------------------------------------
// Kernel 2: persistent sequential scan (the serial bottleneck).
//   pre    = proj_plus[s] + h_{s-1} @ W_h^T          (in-place in rnn[])
//   rnn[s] = (1-ALPHA)*h_{s-1} + ALPHA*relu(pre)
// 8 blocks x 512 threads (16 waves).  Block b owns columns [64b, 64b+64).
// W_h slice staged once in 128 KB LDS, reused for all 1024 steps.
// Wave w: rows 16*(w&7), col groups 2*(w>>3)+{0,1}; A fragment shared.
// Cross-block step sync: monotonic atomic counter + agent fences.
// ---------------------------------------------------------------------------
__global__ __launch_bounds__(512)
void scan_kernel(const float* __restrict__ Wh, float* __restrict__ rnn,
                 int* __restrict__ counter) {
    extern __shared__ float smem[];
    const int lane = threadIdx.x & 31;
    const int wave = threadIdx.x >> 5;            // 0..15
    const int rowg = (wave & 7) * 16;             // batch rows
    const int cg0  = (wave >> 3) * 2;             // col group pair in block
    const int col0 = blockIdx.x * 64 + cg0 * 16;
    const int NK4 = HID / 4;                      // 128

    fill_wfrags(smem, Wh, blockIdx.x * 64, HID, 4, NK4, threadIdx.x, 512);

    for (int s = 0; s < SEQ; ++s) {
        float* cur = rnn + (size_t)s * BATCH * HID;
        const float* prev = (s > 0) ? (cur - (size_t)BATCH * HID) : rnn;

        v8f acc0, acc1;
#pragma unroll
        for (int v = 0; v < 8; ++v) {
            const int m = rowg + ((lane >> 4) << 3) + v;
            acc0[v] = cur[(size_t)m * HID + col0 + (lane & 15)];
            acc1[v] = cur[(size_t)m * HID + col0 + 16 + (lane & 15)];
        }

        if (s > 0) {
            for (int kki = 0; kki < NK4; kki += 4) {   // chunk of 4 fragments
                v2f a[4], b0[4], b1[4];
#pragma unroll
                for (int j = 0; j < 4; ++j) {
                    a[j]  = load_frag(prev, rowg, HID, (kki + j) * 4, lane);
                    b0[j] = lds_frag(smem, cg0,     NK4, kki + j, lane);
                    b1[j] = lds_frag(smem, cg0 + 1, NK4, kki + j, lane);
                }
#pragma unroll
                for (int j = 0; j < 4; ++j) {
                    acc0 = wmma4(a[j], b0[j], acc0);
                    acc1 = wmma4(a[j], b1[j], acc1);
                }
            }
        }

#pragma unroll
        for (int v = 0; v < 8; ++v) {
            const int m = rowg + ((lane >> 4) << 3) + v;
            const size_t i0 = (size_t)m * HID + col0 + (lane & 15);
            const size_t i1 = i0 + 16;
            const float hp0 = (s > 0) ? prev[i0] : 0.0f;
            const float hp1 = (s > 0) ? prev[i1] : 0.0f;
            const float r0 = acc0[v] > 0.0f ? acc0[v] : 0.0f;
            const float r1 = acc1[v] > 0.0f ? acc1[v] : 0.0f;
            cur[i0] = (1.0f - ALPHA_F) * hp0 + ALPHA_F * r0;
            cur[i1] = (1.0f - ALPHA_F) * hp1 + ALPHA_F * r1;
        }

        // ---- grid barrier between steps ----
        __builtin_amdgcn_fence(__ATOMIC_RELEASE, "agent");
        __syncthreads();
        if (threadIdx.x == 0) {
            __hip_atomic_fetch_add(counter, 1, __ATOMIC_ACQ_REL,
                                   __HIP_MEMORY_SCOPE_AGENT);
            const int target = NBLK_SCAN * (s + 1);
            while (__hip_atomic_load(counter, __ATOMIC_ACQUIRE,
                                     __HIP_MEMORY_SCOPE_AGENT) < target) {
                __builtin_amdgcn_s_sleep(2);
            }
        }
        __syncthreads();
        __builtin_amdgcn_fence(__ATOMIC_ACQUIRE, "agent");
    }
}

// ---------------------------------------------------------------------------
// Kernel 3: out[m, o] = rnn[m,:]·W_out[o,:] + b_out[o]
// Block tile 128 x 64 (full OUT), 8 waves, W_out staged in 128 KB LDS.
// Grid: 1024 blocks.
// ---------------------------------------------------------------------------
__global__ __launch_bounds__(256)
void outproj_kernel(const float* __restrict__ act, const float* __restrict__ Wout,
                    const float* __restrict__ bout, float* __restrict__ out) {
    extern __shared__ float smem[];
    const int lane = threadIdx.x & 31;
    const int wave = threadIdx.x >> 5;
    const int rowbase = blockIdx.x * 128 + wave * 16;
    const int NK4 = HID / 4;                     // 128

    fill_wfrags(smem, Wout, 0, HID, 4, NK4, threadIdx.x, 256);

    v8f acc[4];
#pragma unroll
    for (int cg = 0; cg < 4; ++cg)
#pragma unroll
        for (int v = 0; v < 8; ++v) acc[cg][v] = 0.0f;

    for (int kki = 0; kki < NK4; kki += 4) {
        v2f a[4], b[4][4];
#pragma unroll
        for (int j = 0; j < 4; ++j) {
            a[j] = load_frag(act, rowbase, HID, (kki + j) * 4, lane);
#pragma unroll
            for (int cg = 0; cg < 4; ++cg)
                b[j][cg] = lds_frag(smem, cg, NK4, kki + j, lane);
        }
#pragma unroll
        for (int j = 0; j < 4; ++j)
#pragma unroll
            for (int cg = 0; cg < 4; ++cg)
                acc[cg] = wmma4(a[j], b[j][cg], acc[cg]);
    }

#pragma unroll
    for (int cg = 0; cg < 4; ++cg) {
        const int n = cg * 16 + (lane & 15);
        const float bo = bout[n];
#pragma unroll
        for (int v = 0; v < 8; ++v) {
            const int m = rowbase + ((lane >> 4) << 3) + v;
            out[(size_t)m * OUT_DIM + n] = acc[cg][v] + bo;
        }
    }
}

__global__ void init_counter_kernel(int* c) {
    if (threadIdx.x == 0) *c = 0;
}

// ---------------------------------------------------------------------------
// Launch: inputs in setup_inputs() order:
//   0:x  1:noise  2:W_in  3:b_in  4:W_h  5:b_h  6:W_out  7:b_out   (all f32)
// d_out = [ out (MTOT*OUT) | rnn_activity (SEQ*BATCH*HID) ]  (f32)
// ---------------------------------------------------------------------------
extern "C" void kernel_launch(void* const* d_in, const int* in_sizes, int n_in,
                              void* d_out, int out_size, void* d_ws, size_t ws_size,
                              hipStream_t stream) {
    const float* x     = (const float*)d_in[0];
    const float* noise = (const float*)d_in[1];
    const float* Win   = (const float*)d_in[2];
    const float* bin   = (const float*)d_in[3];
    const float* Wh    = (const float*)d_in[4];
    const float* bh    = (const float*)d_in[5];
    const float* Wout  = (const float*)d_in[6];
    const float* bout  = (const float*)d_in[7];

    float* out_ptr = (float*)d_out;
    float* rnn_ptr = out_ptr + (size_t)MTOT * OUT_DIM;
    int*   counter = (int*)d_ws;

    init_counter_kernel<<<1, 64, 0, stream>>>(counter);

    inproj_kernel<<<dim3(MTOT / 128, HID / 64), 256, 32 * 1024, stream>>>(
        x, noise, Win, bin, bh, rnn_ptr);

    scan_kernel<<<NBLK_SCAN, 512, 128 * 1024, stream>>>(Wh, rnn_ptr, counter);

    outproj_kernel<<<MTOT / 128, 256, 128 * 1024, stream>>>(
        rnn_ptr, Wout, bout, out_ptr);
}